// IcosphereConv_10179072491713
// MI455X (gfx1250) — compile-verified
//
#include <hip/hip_runtime.h>

typedef __attribute__((ext_vector_type(2))) float v2f;
typedef __attribute__((ext_vector_type(8))) float v8f;

namespace {
constexpr int kV     = 163842;          // vertices
constexpr int kK     = 6;               // neighbors
constexpr int kG     = 5;               // gradient ops
constexpr int kN     = 2;               // batch
constexpr int kCin   = 32;
constexpr int kCout  = 64;
constexpr int kKdim  = kCin * kG;       // 160 = GEMM reduction dim (c*5+g)
constexpr int kVT    = 16;              // vertices per wave tile (WMMA N)
constexpr int kWaves = 4;               // waves per block
constexpr int kCwF   = kCout * kKdim;   // 10240 floats of conv_w in LDS
constexpr int kGrF   = kVT * kKdim;     // 2560 floats of grads per (wave, n)
}

__global__ __launch_bounds__(kWaves * 32) void icosphere_conv_wmma(
    const float* __restrict__ x,        // (N, Cin, V)
    const int*   __restrict__ nidx,     // (V, K)
    const float* __restrict__ gradw,    // (V, G, K)
    const float* __restrict__ convw,    // (Cout, Cin, G)
    const float* __restrict__ convb,    // (Cout)
    float* __restrict__ out)            // (N, Cout, V)
{
  extern __shared__ float smem[];
  float* cw = smem;                     // [64][160]   A-matrix (shared by all waves)
  float* gr = smem + kCwF;              // [4][2][16][160] per-(wave,n) B staging

  const int tid  = threadIdx.x;
  const int lane = tid & 31;
  const int wave = tid >> 5;
  const int vloc = lane & 15;           // WMMA: M-row (A) / N-col (B,C,D)
  const int half = lane >> 4;           // WMMA: selects K pair {0,1} vs {2,3}

  // Stage conv_w into LDS once per block; its memory layout o*160 + c*5 + g
  // is exactly the GEMM A row layout.
  for (int i = tid; i < kCwF; i += kWaves * 32) cw[i] = convw[i];
  __syncthreads();

  const int vtile = blockIdx.x * kWaves + wave;
  const int v     = vtile * kVT + vloc;           // this lane's vertex
  const int vc    = (v < kV) ? v : (kV - 1);      // clamp for tail-tile loads

  // Per-lane vertex metadata (shared across both batch slices).
  int nb[kK];
  #pragma unroll
  for (int k = 0; k < kK; ++k) nb[k] = nidx[vc * kK + k];
  float gw[kG * kK];
  #pragma unroll
  for (int j = 0; j < kG * kK; ++j) gw[j] = gradw[(size_t)vc * (kG * kK) + j];

  float* grs = gr + wave * (kN * kGrF); // this wave's private [2][16][160] slice

  // ---- Stage 1: grads[n][v][c*5+g] for this wave's 16 vertices --------------
  // lane = (vertex vloc, channel half); each lane covers 16 channels per n.
  #pragma unroll
  for (int n = 0; n < kN; ++n) {
    #pragma unroll 4
    for (int ci = 0; ci < kCin / 2; ++ci) {
      const int c = half * (kCin / 2) + ci;
      const float* xp = x + ((size_t)n * kCin + c) * kV;
      float xv[kK];
      #pragma unroll
      for (int k = 0; k < kK; ++k) xv[k] = xp[nb[k]];   // 6 scattered L2 hits
      #pragma unroll
      for (int g = 0; g < kG; ++g) {
        float s = 0.f;
        #pragma unroll
        for (int k = 0; k < kK; ++k) s = __builtin_fmaf(xv[k], gw[g * kK + k], s);
        grs[n * kGrF + vloc * kKdim + c * kG + g] = s;  // B stored as [v][k]
      }
    }
  }
  // DS ops from the same wave are in-order: the ds_reads below see the
  // ds_writes above without an explicit barrier (wave-private LDS slice).

  // ---- Stage 2: D(64x16) = conv_w(64x160) x grads(160x16), fp32 WMMA --------
  // Batch-fused: each A fragment feeds both n accumulators (8 WMMAs per
  // k-step from 6 ds_read_b64).
  v8f acc[kN][4] = {};                  // [n][M-tile of Cout], 8 VGPRs each
  #pragma unroll 4
  for (int kk = 0; kk < kKdim / 4; ++kk) {
    const int kb = kk * 4 + half * 2;   // this lane supplies K = kb, kb+1
    const v2f b0 = *(const v2f*)(grs + 0 * kGrF + vloc * kKdim + kb);
    const v2f b1 = *(const v2f*)(grs + 1 * kGrF + vloc * kKdim + kb);
    #pragma unroll
    for (int m = 0; m < 4; ++m) {
      const v2f afr = *(const v2f*)(cw + (m * 16 + vloc) * kKdim + kb);
      acc[0][m] = __builtin_amdgcn_wmma_f32_16x16x4_f32(
          false, afr, false, b0, (short)0, acc[0][m], false, false);
      acc[1][m] = __builtin_amdgcn_wmma_f32_16x16x4_f32(
          false, afr, false, b1, (short)0, acc[1][m], false, false);
    }
  }

  // ---- Epilogue: bias + coalesced store (C/D layout: M = 8*half + r, N = vloc)
  if (v < kV) {
    #pragma unroll
    for (int n = 0; n < kN; ++n) {
      #pragma unroll
      for (int m = 0; m < 4; ++m) {
        #pragma unroll
        for (int r = 0; r < 8; ++r) {
          const int o = m * 16 + half * 8 + r;
          out[((size_t)n * kCout + o) * kV + v] = acc[n][m][r] + convb[o];
        }
      }
    }
  }
}

extern "C" void kernel_launch(void* const* d_in, const int* in_sizes, int n_in,
                              void* d_out, int out_size, void* d_ws, size_t ws_size,
                              hipStream_t stream) {
  (void)in_sizes; (void)n_in; (void)out_size; (void)d_ws; (void)ws_size;
  const float* x      = (const float*)d_in[0];
  const int*   nidx   = (const int*)  d_in[1];
  const float* gradw  = (const float*)d_in[2];
  const float* convw  = (const float*)d_in[3];
  const float* convb  = (const float*)d_in[4];
  float*       out    = (float*)d_out;

  const int vtiles = (kV + kVT - 1) / kVT;                 // 10241
  const int blocks = (vtiles + kWaves - 1) / kWaves;       // 2561
  const size_t lds_bytes =
      (size_t)(kCwF + kWaves * kN * kGrF) * sizeof(float); // 120 KB

  icosphere_conv_wmma<<<blocks, kWaves * 32, lds_bytes, stream>>>(
      x, nidx, gradw, convw, convb, out);
}